// DetectionDistillLoss_26104811225326
// MI455X (gfx1250) — compile-verified
//
#include <hip/hip_runtime.h>
#include <hip/hip_bf16.h>
#include <math.h>

#define NP 300
#define NV 100
#define NC 91
#define VPITCH 92            /* LDS row pitch: stride-92 gather is bank-conflict-free */
#define EPSF 1e-7f
#define INV_PI2 0.40528473456935109f /* 4/pi^2 */

typedef __attribute__((ext_vector_type(2))) float v2f;
typedef __attribute__((ext_vector_type(8))) float v8f;

__device__ __forceinline__ float iou_pair(float ax0, float ay0, float ax1, float ay1, float areaA,
                                          float bx0, float by0, float bx1, float by1, float areaB) {
  float lx = fmaxf(ax0, bx0), ly = fmaxf(ay0, by0);
  float rx = fminf(ax1, bx1), ry = fminf(ay1, by1);
  float w = fmaxf(rx - lx, 0.f), h = fmaxf(ry - ly, 0.f);
  float inter = w * h;
  return inter / (areaA + areaB - inter);
}

// ---------------------------------------------------------------------------
// Kernel 1: greedy matching. One block (256 thr) per image. Boxes in LDS.
// ---------------------------------------------------------------------------
__global__ void k_match(const float* __restrict__ pbox, const float* __restrict__ vbox,
                        int* __restrict__ vof, int* __restrict__ mm, float* __restrict__ cnt) {
  const int b = blockIdx.x;
  const int t = threadIdx.x;
  __shared__ float s_pb[NP * 4];
  __shared__ float s_vb[NV * 4];
  __shared__ int s_m[NP];
  __shared__ int s_vof[NP];
  __shared__ int s_pbest[NV];

  for (int i = t; i < NP * 4; i += 256) s_pb[i] = pbox[(size_t)b * NP * 4 + i];
  for (int i = t; i < NV * 4; i += 256) s_vb[i] = vbox[(size_t)b * NV * 4 + i];
  __syncthreads();

  // Phase A: each proxy -> best victim (first-index tie-break via strict >)
  for (int p = t; p < NP; p += 256) {
    float px0 = s_pb[p * 4 + 0], py0 = s_pb[p * 4 + 1];
    float px1 = s_pb[p * 4 + 2], py1 = s_pb[p * 4 + 3];
    float ap = (px1 - px0) * (py1 - py0);
    float best = -1.f; int bi = 0;
    for (int v = 0; v < NV; ++v) {
      float vx0 = s_vb[v * 4 + 0], vy0 = s_vb[v * 4 + 1];
      float vx1 = s_vb[v * 4 + 2], vy1 = s_vb[v * 4 + 3];
      float av = (vx1 - vx0) * (vy1 - vy0);
      float io = iou_pair(vx0, vy0, vx1, vy1, av, px0, py0, px1, py1, ap);
      if (io > best) { best = io; bi = v; }
    }
    s_m[p] = (best >= 0.5f) ? 1 : 0;
    s_vof[p] = bi;
  }

  // Phase B: each victim -> best proxy
  if (t < NV) {
    float vx0 = s_vb[t * 4 + 0], vy0 = s_vb[t * 4 + 1];
    float vx1 = s_vb[t * 4 + 2], vy1 = s_vb[t * 4 + 3];
    float av = (vx1 - vx0) * (vy1 - vy0);
    float best = -1.f; int bp = 0;
    for (int p = 0; p < NP; ++p) {
      float px0 = s_pb[p * 4 + 0], py0 = s_pb[p * 4 + 1];
      float px1 = s_pb[p * 4 + 2], py1 = s_pb[p * 4 + 3];
      float ap = (px1 - px0) * (py1 - py0);
      float io = iou_pair(vx0, vy0, vx1, vy1, av, px0, py0, px1, py1, ap);
      if (io > best) { best = io; bp = p; }
    }
    s_pbest[t] = bp;
  }
  __syncthreads();

  // Phase C: sequential greedy scan over victims (matches jax.lax.scan order)
  if (t == 0) {
    for (int v = 0; v < NV; ++v) {
      int p = s_pbest[v];
      if (!s_m[p]) s_vof[p] = v;
      s_m[p] = 1;
    }
    int c = 0;
    for (int p = 0; p < NP; ++p) c += s_m[p];
    cnt[b] = (float)c;
  }
  __syncthreads();

  for (int p = t; p < NP; p += 256) {
    vof[(size_t)b * NP + p] = s_vof[p];
    mm[(size_t)b * NP + p] = s_m[p];
  }
}

// ---------------------------------------------------------------------------
// Kernel 2: per-image loss. 4 waves/block; coalesced global -> LDS staging;
// strided (WMMA A-layout) gathers served by LDS; V_WMMA_F32_16X16X4_F32 with
// B = ones as an exact fp32 16-row reduction engine.
// ---------------------------------------------------------------------------
__global__ __launch_bounds__(128) void k_loss(
    const float* __restrict__ pbox, const float* __restrict__ plog,
    const float* __restrict__ vbox, const float* __restrict__ vlog,
    const int* __restrict__ vof, const int* __restrict__ mm, float* __restrict__ loss) {
  const int b = blockIdx.x;
  const int tid = threadIdx.x;
  const int lane = tid & 31;
  const int wave = tid >> 5;
  const int r = lane & 15;  // A-matrix row M
  const int h = lane >> 4;  // K half: classes 4k+{0,1} (h=0) / 4k+{2,3} (h=1)
  const float* PL = plog + (size_t)b * NP * NC;
  const float* VL = vlog + (size_t)b * NV * NC;

  __shared__ float s_v[NV * VPITCH];       // 36.8 KB victim logits
  __shared__ float s_p[4][16 * VPITCH];    // 23.6 KB per-wave proxy staging
  __shared__ float s_vmax[NV];
  __shared__ float s_vrs[NV];
  __shared__ float s_w[4];

  // Stage all victim logits for this image (fully coalesced).
  for (int j = tid; j < NV * NC; j += 128) {
    int vv = j / NC, cc = j - vv * NC;
    s_v[vv * VPITCH + cc] = VL[j];
  }
  __syncthreads();

  // Per-victim softmax stats from LDS.
  if (tid < NV) {
    const float* row = s_v + tid * VPITCH;
    float mx = -__builtin_inff();
    for (int c = 0; c < NC; ++c) mx = fmaxf(mx, row[c]);
    float s = 0.f;
    for (int c = 0; c < NC; ++c) s += __expf(row[c] - mx);
    s_vmax[tid] = mx;
    s_vrs[tid] = 1.f / s;
  }
  __syncthreads();

  v8f acc = {0.f, 0.f, 0.f, 0.f, 0.f, 0.f, 0.f, 0.f};
  v2f bones; bones[0] = 1.f; bones[1] = 1.f;
  float loc_sum = 0.f;
  const int NG = (NP + 15) / 16;  // 19
  float* spb = s_p[wave];

  for (int g = wave; g < NG; g += 4) {
    const int base = g * 16;
    if (g + 4 < NG) __builtin_prefetch(PL + (size_t)(base + 64) * NC, 0, 1);

    // Stage this wave's 16 contiguous proxy rows (coalesced); zero-pad tail.
    int lim = (NP - base) * NC;
    if (lim > 16 * NC) lim = 16 * NC;
    for (int j = lane; j < 16 * NC; j += 32) {
      float val = (j < lim) ? PL[(size_t)base * NC + j] : 0.f;
      int rr = j / NC, cc = j - rr * NC;
      spb[rr * VPITCH + cc] = val;
    }
    // Same-wave DS ops are in-order; fence the compiler + wait out DS count.
    asm volatile("s_wait_dscnt 0" ::: "memory");
    __builtin_amdgcn_wave_barrier();

    const int p = base + r;
    const int pe = (p < NP) ? p : (NP - 1);
    const int mk = (p < NP) ? mm[(size_t)b * NP + pe] : 0;
    const int v = vof[(size_t)b * NP + pe];
    const float msk = mk ? 1.f : 0.f;
    const float vmax = s_vmax[v];
    const float vrcp = s_vrs[v] * msk;  // folds match mask -> exact zeros
    const float* prow = spb + r * VPITCH;     // bank-conflict-free gather
    const float* vrow = s_v + v * VPITCH;
    const float tv1 = (h == 0) ? 1.f : 0.f;   // validity of tail slot (class 91 pad)

    // Pass 1: load 46 row values to registers, row max (dup-inclusion is fine).
    float xa[23], xb[23];
    float mx = -__builtin_inff();
#pragma unroll
    for (int k = 0; k < 23; ++k) {
      int c0 = 4 * k + 2 * h;
      int c1 = (c0 + 1 < NC) ? c0 + 1 : NC - 1;  // clamps only at k==22,h==1
      xa[k] = prow[c0];
      xb[k] = prow[c1];
      mx = fmaxf(mx, fmaxf(xa[k], xb[k]));
    }
    mx = fmaxf(mx, __shfl_xor(mx, 16, 32));

    // Pass 2: sum-exp (tail slot gated so class 90 isn't double counted).
    float se = 0.f;
#pragma unroll
    for (int k = 0; k < 23; ++k) {
      se += __expf(xa[k] - mx);
      float e1 = __expf(xb[k] - mx);
      se += (k == 22) ? tv1 * e1 : e1;
    }
    se += __shfl_xor(se, 16, 32);
    const float mxZ = mx + __logf(se);

    // Pass 3: focal distillation contributions -> WMMA row-reduction.
#pragma unroll
    for (int k = 0; k < 23; ++k) {
      int c0 = 4 * k + 2 * h;
      int c1 = (c0 + 1 < NC) ? c0 + 1 : NC - 1;
      float lp0 = xa[k] - mxZ;
      float f0 = 1.f - __expf(lp0);
      float st0 = __expf(vrow[c0] - vmax) * vrcp;
      float a0 = -st0 * f0 * f0 * lp0;
      float lp1 = xb[k] - mxZ;
      float f1 = 1.f - __expf(lp1);
      float st1 = __expf(vrow[c1] - vmax) * vrcp;
      float a1 = -st1 * f1 * f1 * lp1;
      if (k == 22) a1 *= tv1;  // compile-time-selected multiply, no branch
      v2f A; A[0] = a0; A[1] = a1;
      acc = __builtin_amdgcn_wmma_f32_16x16x4_f32(false, A, false, bones,
                                                  (short)0, acc, false, false);
    }

    // CIoU localization loss (h==0 half contributes).
    {
      const float* P = pbox + ((size_t)b * NP + pe) * 4;
      const float* T = vbox + ((size_t)b * NV + v) * 4;
      float p0 = P[0], p1 = P[1], p2 = P[2], p3 = P[3];
      float t0 = T[0], t1 = T[1], t2 = T[2], t3 = T[3];
      float pw = p2 - p0, ph = p3 - p1, tw = t2 - t0, th = t3 - t1;
      float iw = fmaxf(fminf(p2, t2) - fmaxf(p0, t0), 0.f);
      float ih = fmaxf(fminf(p3, t3) - fmaxf(p1, t1), 0.f);
      float inter = iw * ih;
      float uni = pw * ph + tw * th - inter + EPSF;
      float iou = inter / uni;
      float cw = fmaxf(p2, t2) - fminf(p0, t0);
      float ch = fmaxf(p3, t3) - fminf(p1, t1);
      float c2 = cw * cw + ch * ch + EPSF;
      float dx = (p0 + p2) * 0.5f - (t0 + t2) * 0.5f;
      float dy = (p1 + p3) * 0.5f - (t1 + t3) * 0.5f;
      float rho2 = dx * dx + dy * dy;
      float dth = atanf(tw / (th + EPSF)) - atanf(pw / (ph + EPSF));
      float vv = INV_PI2 * dth * dth;
      float alpha = vv / (1.f - iou + vv + EPSF);
      float ciou = iou - rho2 / c2 - alpha * vv;
      loc_sum += ((h == 0) ? msk : 0.f) * (1.f - ciou);
    }
  }

  // D holds row sums broadcast across 16 columns -> /16 is exact.
  float tsum = (acc[0] + acc[1] + acc[2] + acc[3] + acc[4] + acc[5] + acc[6] + acc[7])
                   * 0.0625f + loc_sum;
  for (int o = 16; o >= 1; o >>= 1) tsum += __shfl_xor(tsum, o, 32);

  if (lane == 0) s_w[wave] = tsum;
  __syncthreads();
  if (tid == 0) {
    float tot = 0.f;
    for (int w = 0; w < 4; ++w) tot += s_w[w];
    loss[b] = tot;
  }
}

// ---------------------------------------------------------------------------
// Kernel 3: deterministic final reduction + division.
// ---------------------------------------------------------------------------
__global__ void k_final(const float* __restrict__ lossb, const float* __restrict__ cntb,
                        float* __restrict__ out, int nb) {
  __shared__ float sl[256];
  __shared__ float sc[256];
  const int t = threadIdx.x;
  float l = 0.f, c = 0.f;
  for (int i = t; i < nb; i += 256) { l += lossb[i]; c += cntb[i]; }
  sl[t] = l; sc[t] = c;
  __syncthreads();
  for (int s = 128; s > 0; s >>= 1) {
    if (t < s) { sl[t] += sl[t + s]; sc[t] += sc[t + s]; }
    __syncthreads();
  }
  if (t == 0) out[0] = sl[0] / sc[0];
}

extern "C" void kernel_launch(void* const* d_in, const int* in_sizes, int n_in,
                              void* d_out, int out_size, void* d_ws, size_t ws_size,
                              hipStream_t stream) {
  const float* pbox = (const float*)d_in[0];
  const float* plog = (const float*)d_in[3];
  const float* vbox = (const float*)d_in[4];
  const float* vlog = (const float*)d_in[7];
  const int B = in_sizes[0] / (NP * 4);

  char* w = (char*)d_ws;
  int* vof = (int*)w;       w += (size_t)B * NP * sizeof(int);
  int* mm = (int*)w;        w += (size_t)B * NP * sizeof(int);
  float* lossb = (float*)w; w += (size_t)B * sizeof(float);
  float* cntb = (float*)w;  w += (size_t)B * sizeof(float);

  k_match<<<B, 256, 0, stream>>>(pbox, vbox, vof, mm, cntb);
  k_loss<<<B, 128, 0, stream>>>(pbox, plog, vbox, vlog, vof, mm, lossb);
  k_final<<<1, 256, 0, stream>>>(lossb, cntb, (float*)d_out, B);
}